// RBFKernel_65481071406641
// MI455X (gfx1250) — compile-verified
//
#include <hip/hip_runtime.h>

typedef __attribute__((ext_vector_type(2))) float v2f;
typedef __attribute__((ext_vector_type(8))) float v8f;

#define RBF_GAMMA 1.0f
#define LOG2E     1.4426950408889634f

static constexpr int NROW = 8192;   // N (rows of x)
static constexpr int MROW = 8192;   // M (rows of y)
static constexpr int DDIM = 64;     // feature dim

// ---------------------------------------------------------------------------
// Pass 1: per-row ||x_i||^2, ||y_i||^2 and diagonal normalizer
//   s[i] = 1/sqrt(diag_i) = exp(+0.5*gamma*max(||x_i - y_i||^2, 0))
// ---------------------------------------------------------------------------
__global__ void rbf_rowstats(const float* __restrict__ x,
                             const float* __restrict__ y,
                             float* __restrict__ x2,
                             float* __restrict__ y2,
                             float* __restrict__ s) {
  int i = blockIdx.x * blockDim.x + threadIdx.x;
  if (i >= NROW) return;
  const float4* xr = (const float4*)(x + (size_t)i * DDIM);
  const float4* yr = (const float4*)(y + (size_t)i * DDIM);
  float sx = 0.0f, sy = 0.0f, sxy = 0.0f;
#pragma unroll
  for (int q = 0; q < DDIM / 4; ++q) {
    float4 a = xr[q];
    float4 b = yr[q];
    sx  = fmaf(a.x, a.x, fmaf(a.y, a.y, fmaf(a.z, a.z, fmaf(a.w, a.w, sx))));
    sy  = fmaf(b.x, b.x, fmaf(b.y, b.y, fmaf(b.z, b.z, fmaf(b.w, b.w, sy))));
    sxy = fmaf(a.x, b.x, fmaf(a.y, b.y, fmaf(a.z, b.z, fmaf(a.w, b.w, sxy))));
  }
  x2[i] = sx;
  y2[i] = sy;
  float d2 = fmaxf(sx + sy - 2.0f * sxy, 0.0f);
  // exp(0.5*gamma*d2) via v_exp_f32
  s[i] = __builtin_amdgcn_exp2f(0.5f * RBF_GAMMA * LOG2E * d2);
}

// ---------------------------------------------------------------------------
// Pass 2: one wave32 per 16x16 output tile; 8 waves/block -> 32x64 patch.
// C tile = x[i0:i0+16, :] @ y[j0:j0+16, :]^T via 16 x V_WMMA_F32_16X16X4_F32.
//
// f32 WMMA fragment layouts (ISA 7.12.2):
//   A (16x4):  lane L holds row m=L&15; VGPR0/1 = K = 4*kk + 2*(L>>4) (+1)
//   B (4x16):  lane L holds col n=L&15; VGPR0/1 = K = 4*kk + 2*(L>>4) (+1)
//   C/D (16x16): VGPR r -> row r + 8*(L>>4), col L&15
// Both fragments are contiguous float2 loads from row-major x / y.
// ---------------------------------------------------------------------------
__global__ __launch_bounds__(256) void rbf_wmma_tile(
    const float* __restrict__ x, const float* __restrict__ y,
    const float* __restrict__ x2, const float* __restrict__ y2,
    const float* __restrict__ s, float* __restrict__ out) {
  const int lane = threadIdx.x & 31;
  const int wave = threadIdx.x >> 5;       // 0..7
  const int n  = lane & 15;                // tile col (B/C) or tile row (A)
  const int kh = lane >> 4;                // K-pair half / C row-half

  const int i0 = (blockIdx.y * 2 + (wave >> 2)) * 16;  // 32 rows / block
  const int j0 = (blockIdx.x * 4 + (wave & 3)) * 16;   // 64 cols / block

  const float* arow = x + (size_t)(i0 + n) * DDIM + 2 * kh;
  const float* brow = y + (size_t)(j0 + n) * DDIM + 2 * kh;

  v8f c = {};
#pragma unroll
  for (int kk = 0; kk < DDIM / 4; ++kk) {
    v2f a = *(const v2f*)(arow + kk * 4);
    v2f b = *(const v2f*)(brow + kk * 4);
    // (neg_a, A, neg_b, B, c_mod, C, reuse_a, reuse_b)
    c = __builtin_amdgcn_wmma_f32_16x16x4_f32(
        false, a, false, b, (short)0, c, false, false);
  }

  // Epilogue: d2 = x2[i] + y2[j] - 2*dot, clamped; out = exp(-g*d2)*s[i]*s[j]
  const float y2n = y2[j0 + n];
  const float sn  = s[j0 + n];
  const int rowbase = i0 + kh * 8;

#pragma unroll
  for (int r = 0; r < 8; ++r) {
    const int i = rowbase + r;
    float d2 = fmaxf(x2[i] + y2n - 2.0f * c[r], 0.0f);
    float v  = __builtin_amdgcn_exp2f(-RBF_GAMMA * LOG2E * d2) * s[i] * sn;
    out[(size_t)i * MROW + j0 + n] = v;
  }
}

// ---------------------------------------------------------------------------
extern "C" void kernel_launch(void* const* d_in, const int* in_sizes, int n_in,
                              void* d_out, int out_size, void* d_ws, size_t ws_size,
                              hipStream_t stream) {
  const float* x = (const float*)d_in[0];   // [8192, 64] fp32
  const float* y = (const float*)d_in[1];   // [8192, 64] fp32
  float* out = (float*)d_out;               // [8192, 8192] fp32

  float* x2 = (float*)d_ws;                 // N floats
  float* y2 = x2 + NROW;                    // M floats
  float* s  = y2 + MROW;                    // N floats (needs 96 KB total)

  // Pass 1: row stats
  rbf_rowstats<<<dim3((NROW + 255) / 256), dim3(256), 0, stream>>>(x, y, x2, y2, s);

  // Pass 2: WMMA tiles. Block covers 32 rows x 64 cols.
  dim3 grid(MROW / 64, NROW / 32);
  rbf_wmma_tile<<<grid, dim3(256), 0, stream>>>(x, y, x2, y2, s, out);
}